// memory_Block_31585189494936
// MI455X (gfx1250) — compile-verified
//
#include <hip/hip_runtime.h>
#include <math.h>

typedef __attribute__((ext_vector_type(2))) float v2f;
typedef __attribute__((ext_vector_type(8))) float v8f;
typedef __attribute__((ext_vector_type(4))) unsigned int u32x4;
typedef __attribute__((ext_vector_type(4))) int i32x4;
typedef __attribute__((ext_vector_type(8))) int i32x8;

constexpr int NN = 8192;   // rows of q (= b*h*w)
constexpr int MM = 8192;   // memory slots
constexpr int CC = 256;    // channels

// ---------------------------------------------------------------- zero ws
__global__ __launch_bounds__(256) void k_zero(float* __restrict__ p, int n) {
  int i = blockIdx.x * 256 + threadIdx.x;
  if (i < n) p[i] = 0.0f;
}

// ------------------------------------------------- mem (C,M) -> memT (M,C)
// TDM (tensor_load_to_lds) DMAs a 32x32 f32 tile into LDS with 1-DWORD row
// padding (stride 33 -> conflict-free transposed reads); threads then write
// the transposed tile coalesced.
__global__ __launch_bounds__(256) void k_transpose(const float* __restrict__ mem,
                                                   float* __restrict__ memT) {
  __shared__ float tile[32 * 33 + 8];
  const int mb = blockIdx.x * 32;
  const int cb = blockIdx.y * 32;
  const int tid = threadIdx.x;

  if (tid < 32) {  // wave 0 issues the DMA (TDM ignores EXEC; one issue/wave)
    unsigned long long ga =
        (unsigned long long)(const void*)(mem + (size_t)cb * MM + mb);
    unsigned int lds = (unsigned int)(unsigned long long)(void*)&tile[0];

    // D# group 0: count=1 | lds_addr | global_addr[56:0] | type=2
    u32x4 g0;
    g0.x = 1u;                                    // count=1, user mode
    g0.y = lds;                                   // lds_addr (bytes)
    g0.z = (unsigned int)(ga & 0xffffffffull);    // global_addr[31:0]
    g0.w = (unsigned int)((ga >> 32) & 0x1ffffffull) | (2u << 30);  // type=2

    // D# group 1: data_size=4B(code2), pad_enable, pad_interval=32DW(code4),
    // pad_amount=1DW(code0); tensor 8192x8192, tile 32x32, dim0 stride 8192.
    i32x8 g1;
    g1[0] = (int)((2u << 16) | (1u << 20) | (4u << 22));      // flags
    g1[1] = (int)((8192u & 0xffffu) << 16);                   // tensor_dim0 lo16
    g1[2] = (int)((8192u >> 16) | ((8192u & 0xffffu) << 16)); // dim0 hi / dim1 lo
    g1[3] = (int)((8192u >> 16) | (32u << 16));               // dim1 hi / tile_dim0
    g1[4] = (int)(32u);                                       // tile_dim1 (tile_dim2=0)
    g1[5] = (int)(8192u);                                     // tensor_dim0_stride lo32
    g1[6] = 0;                                                // stride hi / dim1_stride lo
    g1[7] = 0;

    i32x4 z4 = {0, 0, 0, 0};
#if __clang_major__ >= 23
    i32x8 z8 = {0, 0, 0, 0, 0, 0, 0, 0};
    __builtin_amdgcn_tensor_load_to_lds(g0, g1, z4, z4, z8, 0);
#else
    __builtin_amdgcn_tensor_load_to_lds(g0, g1, z4, z4, 0);
#endif
    __builtin_amdgcn_s_wait_tensorcnt(0);
  }
  __syncthreads();

  // tile[j*33 + i] = mem[cb+j][mb+i]; write memT[mb+i][cb+j], coalesced in j
  const int c = tid & 31;   // j (channel within tile)
  const int i0 = tid >> 5;  // m within tile, 8 threads -> 4 iters
#pragma unroll
  for (int k = 0; k < 4; ++k) {
    int i = i0 + k * 8;
    memT[(size_t)(mb + i) * CC + cb + c] = tile[c * 33 + i];
  }
}

// ---------------------------------------------------------------- pass A
// One WG = 16 q rows. 8 waves split the 512 column tiles (16 cols each).
// Sweep 1: exact softmax stats via WMMA score tiles.
// Sweep 2: recompute scores, p = s_mem, WMMA-accumulate E_hat, fused
//          per-column argmax via packed u64 atomicMax.
__global__ __launch_bounds__(256) void k_attn(const float* __restrict__ q,
                                              const float* __restrict__ memT,
                                              float* __restrict__ out,
                                              float* __restrict__ rowOff,
                                              unsigned long long* __restrict__ colBest) {
  __shared__ float qs2[16 * 256];      // paired A layout: [(k>>1)*32 + r*2 + (k&1)]
  __shared__ float ehat[16 * 257];     // padded E_hat accumulator
  __shared__ float ps[8][16 * 17];     // per-wave P transpose staging (padded)
  __shared__ float sMw[8][16], sLw[8][16];
  __shared__ float sM[16], sInvL[16], sOff[16];

  const int tid  = threadIdx.x;
  const int wave = tid >> 5;
  const int lane = tid & 31;
  const int hf   = lane >> 4;      // lane half (wave32)
  const int l16  = lane & 15;
  const int rowbase = blockIdx.x * 16;

  for (int idx = tid; idx < 16 * 256; idx += 256) {
    int r = idx >> 8, k = idx & 255;
    qs2[(k >> 1) * 32 + r * 2 + (k & 1)] = q[(rowbase + r) * CC + k];
  }
  for (int idx = tid; idx < 16 * 257; idx += 256) ehat[idx] = 0.0f;
  __syncthreads();

  // ---------------- sweep 1: row max / sumexp ----------------
  float run_m[8], run_l[8];
#pragma unroll
  for (int j = 0; j < 8; ++j) { run_m[j] = -3.0e38f; run_l[j] = 0.0f; }

  for (int it = 0; it < 64; ++it) {
    const int colbase = (wave + (it << 3)) << 4;
    const float* bptr = memT + (colbase + l16) * CC + hf * 2;
    if (it < 63) __builtin_prefetch((const void*)(bptr + 128 * CC), 0, 1);
    v8f c8 = {0, 0, 0, 0, 0, 0, 0, 0};
#pragma unroll
    for (int k0 = 0; k0 < 256; k0 += 4) {
      v2f a = *(const v2f*)(qs2 + ((k0 + 2 * hf) >> 1) * 32 + l16 * 2);
      v2f b = *(const v2f*)(bptr + k0);
      c8 = __builtin_amdgcn_wmma_f32_16x16x4_f32(false, a, false, b, (short)0, c8,
                                                 false, false);
    }
#pragma unroll
    for (int j = 0; j < 8; ++j) {
      float v = c8[j];
      float tm = v;
      tm = fmaxf(tm, __shfl_xor(tm, 1, 32));
      tm = fmaxf(tm, __shfl_xor(tm, 2, 32));
      tm = fmaxf(tm, __shfl_xor(tm, 4, 32));
      tm = fmaxf(tm, __shfl_xor(tm, 8, 32));
      float nm = fmaxf(run_m[j], tm);
      float e = __expf(v - nm);
      e += __shfl_xor(e, 1, 32);
      e += __shfl_xor(e, 2, 32);
      e += __shfl_xor(e, 4, 32);
      e += __shfl_xor(e, 8, 32);
      run_l[j] = run_l[j] * __expf(run_m[j] - nm) + e;
      run_m[j] = nm;
    }
  }
  if (l16 == 0) {
#pragma unroll
    for (int j = 0; j < 8; ++j) { sMw[wave][j + 8 * hf] = run_m[j]; sLw[wave][j + 8 * hf] = run_l[j]; }
  }
  __syncthreads();
  if (tid < 16) {
    float M = -3.0e38f;
    for (int w = 0; w < 8; ++w) M = fmaxf(M, sMw[w][tid]);
    float L = 0.0f;
    for (int w = 0; w < 8; ++w) L += sLw[w][tid] * __expf(sMw[w][tid] - M);
    sM[tid] = M;
    sInvL[tid] = 1.0f / L;
    float off = M + __logf(L);
    sOff[tid] = off;
    rowOff[rowbase + tid] = off;
  }
  __syncthreads();

  // ---------------- sweep 2: p = s_mem, E_hat WMMA, fused column argmax --
  float fm[8], fil[8];
#pragma unroll
  for (int j = 0; j < 8; ++j) { int r = j + 8 * hf; fm[j] = sM[r]; fil[j] = sInvL[r]; }

  v8f acc[16];
#pragma unroll
  for (int cc = 0; cc < 16; ++cc) acc[cc] = (v8f){0, 0, 0, 0, 0, 0, 0, 0};

  float* myps = &ps[wave][0];

  for (int it = 0; it < 64; ++it) {
    const int colbase = (wave + (it << 3)) << 4;
    const float* bptr = memT + (colbase + l16) * CC + hf * 2;
    if (it < 63) __builtin_prefetch((const void*)(bptr + 128 * CC), 0, 1);
    v8f c8 = {0, 0, 0, 0, 0, 0, 0, 0};
#pragma unroll
    for (int k0 = 0; k0 < 256; k0 += 4) {
      v2f a = *(const v2f*)(qs2 + ((k0 + 2 * hf) >> 1) * 32 + l16 * 2);
      v2f b = *(const v2f*)(bptr + k0);
      c8 = __builtin_amdgcn_wmma_f32_16x16x4_f32(false, a, false, b, (short)0, c8,
                                                 false, false);
    }
    float bestv = -1.0f;
    int bestr = 0;
#pragma unroll
    for (int j = 0; j < 8; ++j) {
      float p = __expf(c8[j] - fm[j]) * fil[j];   // exact s_mem value
      if (p > bestv) { bestv = p; bestr = j + 8 * hf; }
      myps[(j + 8 * hf) * 17 + l16] = p;          // stage P for transpose
    }
    // combine lane halves: lane l and l^16 hold the same column
    float ov = __shfl_xor(bestv, 16, 32);
    int orr = __shfl_xor(bestr, 16, 32);
    if (ov > bestv) { bestv = ov; bestr = orr; }
    if (hf == 0) {
      unsigned long long key =
          (((unsigned long long)__float_as_uint(bestv)) << 32) |
          (unsigned int)(rowbase + bestr);
      atomicMax(colBest + colbase + l16, key);
    }
    __syncthreads();   // P staged (uniform trip count across all 8 waves)

    // E_hat tile accumulate: acc[cc] += P(16x16) @ memT[colbase:colbase+16, :]
    const float* bp2 = memT + (colbase + (hf << 1)) * CC;
#pragma unroll
    for (int cc = 0; cc < 16; ++cc) {
      int c = cc * 16 + l16;
#pragma unroll
      for (int k0 = 0; k0 < 16; k0 += 4) {
        v2f a;
        a.x = myps[l16 * 17 + k0 + 2 * hf];
        a.y = myps[l16 * 17 + k0 + 2 * hf + 1];
        v2f b;
        b.x = bp2[k0 * CC + c];
        b.y = bp2[(k0 + 1) * CC + c];
        acc[cc] = __builtin_amdgcn_wmma_f32_16x16x4_f32(false, a, false, b, (short)0,
                                                        acc[cc], false, false);
      }
    }
  }
  __syncthreads();

  // merge per-wave partial E_hat into shared accumulator
#pragma unroll
  for (int cc = 0; cc < 16; ++cc) {
    int c = cc * 16 + l16;
#pragma unroll
    for (int j = 0; j < 8; ++j) atomicAdd(&ehat[(j + 8 * hf) * 257 + c], acc[cc][j]);
  }
  __syncthreads();

  // write E_out: out[b, ch, h, w]; n = rowbase..rowbase+15 share b and hw base
  const int bb = rowbase >> 10;
  const int hw0 = rowbase & 1023;
  float* outb = out + bb * (512 * 1024) + hw0;
  const int r = tid & 15;
  for (int c0 = tid >> 4; c0 < 256; c0 += 16) {
    float qv = qs2[(c0 >> 1) * 32 + r * 2 + (c0 & 1)];
    outb[c0 * 1024 + r] = qv;                         // q part (ch < 256)
    outb[(256 + c0) * 1024 + r] = ehat[r * 257 + c0]; // E_hat part
  }
}

// ---------------------------------------------------------------- pass C
// vals[i] = s_mem[i, k_i[i]] exactly; scatter-add vals[i]*q[i] into sum_wq.
__global__ __launch_bounds__(256) void k_scatter(const float* __restrict__ q,
                                                 const float* __restrict__ memT,
                                                 const float* __restrict__ rowOff,
                                                 const unsigned long long* __restrict__ colBest,
                                                 float* __restrict__ sum_wq) {
  const int rrow = blockIdx.x * 8 + (threadIdx.x >> 5);  // one wave per row
  const int lane = threadIdx.x & 31;
  const int kk = (int)(unsigned int)(colBest[rrow] & 0xffffffffull);
  const float* qp = q + rrow * CC;
  const float* mp = memT + kk * CC;
  float s = 0.0f;
#pragma unroll
  for (int i = 0; i < 8; ++i) s += qp[lane + i * 32] * mp[lane + i * 32];
  s += __shfl_xor(s, 1, 32);
  s += __shfl_xor(s, 2, 32);
  s += __shfl_xor(s, 4, 32);
  s += __shfl_xor(s, 8, 32);
  s += __shfl_xor(s, 16, 32);
  const float val = __expf(s - rowOff[rrow]);
  float* dst = sum_wq + kk * CC;
#pragma unroll
  for (int i = 0; i < 8; ++i) atomicAdd(&dst[lane + i * 32], val * qp[lane + i * 32]);
}

// ---------------------------------------------------------------- pass D
// upd[c][m] = normalize_c(memT[m][c] + sum_wq[m][c]); coalesced (C,M) writes.
__global__ __launch_bounds__(256) void k_norm(const float* __restrict__ memT,
                                              const float* __restrict__ sum_wq,
                                              float* __restrict__ outU) {
  __shared__ float us[16][257];
  __shared__ float nrm2[16];
  const int mbase = blockIdx.x * 16;
  const int t = threadIdx.x;
  if (t < 16) nrm2[t] = 0.0f;
  __syncthreads();
  const int c = t;  // 256 threads == 256 channels, coalesced reads
#pragma unroll
  for (int i = 0; i < 16; ++i) {
    float v = memT[(mbase + i) * CC + c] + sum_wq[(mbase + i) * CC + c];
    us[i][c] = v;
    float p2 = v * v;
    p2 += __shfl_xor(p2, 1, 32);
    p2 += __shfl_xor(p2, 2, 32);
    p2 += __shfl_xor(p2, 4, 32);
    p2 += __shfl_xor(p2, 8, 32);
    p2 += __shfl_xor(p2, 16, 32);
    if ((t & 31) == 0) atomicAdd(&nrm2[i], p2);
  }
  __syncthreads();
  const int ml = t & 15;
  const int cg = t >> 4;
  const float sc = 1.0f / fmaxf(sqrtf(nrm2[ml]), 1e-12f);
#pragma unroll
  for (int ci = 0; ci < 16; ++ci) {
    int cidx = cg + ci * 16;
    outU[cidx * MM + mbase + ml] = us[ml][cidx] * sc;  // contiguous per 16 lanes
  }
}

// ---------------------------------------------------------------- launch
extern "C" void kernel_launch(void* const* d_in, const int* in_sizes, int n_in,
                              void* d_out, int out_size, void* d_ws, size_t ws_size,
                              hipStream_t stream) {
  const float* E = (const float*)d_in[0];     // (8,32,32,256) == q (8192,256)
  const float* mem = (const float*)d_in[1];   // (256,8192)
  float* out = (float*)d_out;                 // E_out (4194304) then upd (2097152)

  char* ws = (char*)d_ws;
  float* memT = (float*)ws;                                        // 8 MB
  float* sum_wq = (float*)(ws + (size_t)(8u << 20));               // 8 MB
  unsigned long long* colBest =
      (unsigned long long*)(ws + (size_t)(16u << 20));             // 64 KB
  float* rowOff = (float*)(ws + (size_t)(16u << 20) + 65536u);     // 32 KB

  // zero sum_wq (2097152 f32) + colBest (16384 f32-equivalent), contiguous
  k_zero<<<8256, 256, 0, stream>>>(sum_wq, 2097152 + 16384);
  k_transpose<<<dim3(MM / 32, CC / 32), 256, 0, stream>>>(mem, memT);
  k_attn<<<NN / 16, 256, 0, stream>>>(E, memT, out, rowOff, colBest);
  k_scatter<<<NN / 32 / 8, 256, 0, stream>>>(E, memT, rowOff, colBest, sum_wq);
  k_norm<<<MM / 16, 256, 0, stream>>>(memT, sum_wq, out + 4194304);
}